// Qatten_28149215658642
// MI455X (gfx1250) — compile-verified
//
#include <hip/hip_runtime.h>
#include <hip/hip_bf16.h>

typedef __attribute__((ext_vector_type(2))) float v2f;
typedef __attribute__((ext_vector_type(8))) float v8f;

#define N_AGENTS 16
#define UNIT_DIM 64
#define STATE    1024
#define EDIM     128
#define HEADS    8
#define BSZ      16384
#define REG_COEF 0.001f
#define SCALE_F  0.08838834764831845f   // 1/sqrt(128)
#define MASK_F   -99999999.0f

#define ROWS         32     // batch rows per block (2 groups of 16)
#define GROUPS       2
#define NT_PER_WAVE  9      // 72 N-tiles (64 sel + 8 v1) / 8 waves
#define SEL_HSTRIDE  132    // 128 + 4 pad (bank-conflict mitigation)
#define SEL_RSTRIDE  (HEADS * SEL_HSTRIDE)   // 1056 floats per batch row
#define M_STRIDE     66     // 65 + 1 pad

__launch_bounds__(256)
__global__ void qatten_fused(const float* __restrict__ q_values,
                             const float* __restrict__ states,
                             const int*   __restrict__ actions,
                             const float* __restrict__ W_sel,
                             const float* __restrict__ W_key,
                             const float* __restrict__ W_v1,
                             const float* __restrict__ b_v1,
                             const float* __restrict__ W_v2,
                             const float* __restrict__ b_v2,
                             const float* __restrict__ W_wh,
                             const float* __restrict__ b_wh,
                             float* __restrict__ out,
                             float* __restrict__ ws)
{
    __shared__ float sSel[GROUPS * 16 * SEL_RSTRIDE];   // 132 KB: sel (h,e), both groups
    __shared__ float sM  [8 * HEADS * M_STRIDE];        // 16.9 KB: per-wave m[h][u]
    __shared__ float sWh [ROWS * HEADS];                //  1.0 KB: |w_head|
    __shared__ float sEntW[8 * HEADS];
    __shared__ float sRegW[8];

    const int tid  = threadIdx.x;
    const int wv   = tid >> 5;        // wave 0..7 (wave32)
    const int lane = tid & 31;
    const int b0   = blockIdx.x * ROWS;

    // ---------- Phase 0: w_head = |states @ W_wh^T + b_wh|  (one dot per thread) ----------
    {
        const int r = tid >> 3, h = tid & 7;       // 32 rows x 8 heads = 256 threads
        const float* xr = states + (size_t)(b0 + r) * STATE;
        const float* wr = W_wh + (size_t)h * STATE;
        float s = 0.f;
        #pragma unroll 8
        for (int j = 0; j < STATE; ++j) s = fmaf(xr[j], wr[j], s);
        sWh[r * HEADS + h] = fabsf(s + b_wh[h]);
    }
    // consumed after the single __syncthreads below

    // ---------- Phase 1: f32 WMMA GEMM  D(32x1152) = X(32x1024) x [W_sel|W_v1]^T ----------
    // A frag (16x4 f32): lanes 0-15 rows, VGPR0/1 = K {kk, kk+1}, kk = (lane>>4)*2
    // B frag (4x16 f32): lane n = lane&15, same kk pattern; B[k][n] = W[n][k]
    // Two 16-row accumulator groups share every B fragment (2 wmma per B load).
    const int nn = lane & 15;
    const int kk = (lane >> 4) * 2;
    const float* aptr0 = states + (size_t)(b0 + nn) * STATE + kk;
    const float* aptr1 = states + (size_t)(b0 + 16 + nn) * STATE + kk;

    const float* bptr[NT_PER_WAVE];
    #pragma unroll
    for (int t = 0; t < NT_PER_WAVE; ++t) {
        const int T = wv * NT_PER_WAVE + t;
        const int g = T * 16 + nn;
        const float* wrow = (T < 64) ? (W_sel + (size_t)g * STATE)
                                     : (W_v1 + (size_t)(g - 1024) * STATE);
        bptr[t] = wrow + kk;
    }

    v8f acc[GROUPS][NT_PER_WAVE];
    #pragma unroll
    for (int g = 0; g < GROUPS; ++g)
        #pragma unroll
        for (int t = 0; t < NT_PER_WAVE; ++t) acc[g][t] = (v8f)0.f;

    for (int k0 = 0; k0 < STATE; k0 += 4) {
        v2f a0 = *(const v2f*)(aptr0 + k0);
        v2f a1 = *(const v2f*)(aptr1 + k0);
        #pragma unroll
        for (int t = 0; t < NT_PER_WAVE; ++t) {
            v2f b = *(const v2f*)(bptr[t] + k0);   // shared by both row groups
            acc[0][t] = __builtin_amdgcn_wmma_f32_16x16x4_f32(
                            false, a0, false, b, (short)0, acc[0][t], false, false);
            acc[1][t] = __builtin_amdgcn_wmma_f32_16x16x4_f32(
                            false, a1, false, b, (short)0, acc[1][t], false, false);
        }
    }

    // ---------- Wave 7: V-net entirely in registers (owns all h1 tiles T=64..71) ----------
    // v[r] = relu(h1[r] + b_v1) . W_v2 + b_v2 ; h1 row r lives in half-wave r>>3, slot r&7.
    if (wv == 7) {
        float wv2c[8], bv1c[8];
        #pragma unroll
        for (int t = 1; t <= 8; ++t) {
            const int e = (t - 1) * 16 + nn;
            wv2c[t - 1] = W_v2[e];
            bv1c[t - 1] = b_v1[e];
        }
        const float bv2 = b_v2[0];
        #pragma unroll
        for (int g = 0; g < GROUPS; ++g) {
            #pragma unroll
            for (int r = 0; r < 16; ++r) {
                float s = 0.f;
                if ((lane >> 4) == (r >> 3)) {
                    #pragma unroll
                    for (int t = 1; t <= 8; ++t) {
                        const float hv = acc[g][t][r & 7] + bv1c[t - 1];
                        s = fmaf(fmaxf(hv, 0.f), wv2c[t - 1], s);
                    }
                }
                #pragma unroll
                for (int off = 16; off >= 1; off >>= 1)
                    s += __shfl_xor(s, off, 32);
                if (lane == 0) out[BSZ + b0 + g * 16 + r] = s + bv2;
            }
        }
    }

    // ---------- Drain ALL sel fragments to double-buffered LDS (no liveness past here) ----
    // C layout: lanes 0-15: M = v, N = lane; lanes 16-31: M = v + 8, N = lane-16
    #pragma unroll
    for (int g = 0; g < GROUPS; ++g) {
        #pragma unroll
        for (int t = 0; t < NT_PER_WAVE; ++t) {
            const int T = wv * NT_PER_WAVE + t;
            if (T < 64) {
                const int n = T * 16 + nn;
                const int h = n >> 7, e = n & 127;
                #pragma unroll
                for (int v = 0; v < 8; ++v) {
                    const int m = v + ((lane >> 4) << 3);
                    sSel[(g * 16 + m) * SEL_RSTRIDE + h * SEL_HSTRIDE + e] = acc[g][t][v];
                }
            }
        }
    }
    __syncthreads();   // sSel (both groups) + sWh visible to all waves

    // ---------- Phase 2: fused attention for all 4 rows owned by this wave ----------
    float racc = 0.f;                 // sum of raw logits^2
    float eacc[4] = {0.f, 0.f, 0.f, 0.f};
    const int halfsel = lane >> 4;    // 0/1: head parity owned by this half-wave
    float* mbuf = &sM[wv * (HEADS * M_STRIDE)];

    for (int g = 0; g < GROUPS; ++g) {
        for (int rr = 0; rr < 2; ++rr) {
            const int rl = wv * 2 + rr;                  // local row 0..15
            const int rg = g * 16 + rl;                  // row within block 0..31
            const size_t brow = (size_t)(b0 + rg);
            const float* selg = &sSel[g * 16 * SEL_RSTRIDE];

            // m[h][u] = sum_e sel[r][h][e] * W_key[h][e][u]   (520 dots of length 128)
            for (int idx = lane; idx < HEADS * (UNIT_DIM + 1); idx += 32) {
                const int h = idx / (UNIT_DIM + 1);
                const int u = idx - h * (UNIT_DIM + 1);
                const float* selrow = &selg[rl * SEL_RSTRIDE + h * SEL_HSTRIDE];
                const float* wk = W_key + (size_t)h * EDIM * (UNIT_DIM + 1) + u;
                float s = 0.f;
                #pragma unroll 8
                for (int e = 0; e < EDIM; ++e)
                    s = fmaf(selrow[e], wk[e * (UNIT_DIM + 1)], s);
                mbuf[h * M_STRIDE + u] = s;
            }
            // cross-lane LDS handoff within the wave: drain DS counter explicitly
            asm volatile("s_wait_dscnt 0" ::: "memory");

            const int   a   = lane & 15;
            const float qv  = q_values[brow * N_AGENTS + a];
            const int   act = actions [brow * N_AGENTS + a];
            const float* urow = states + brow * STATE + a * UNIT_DIM;

            float hqv[4];
            #pragma unroll
            for (int i = 0; i < 4; ++i) {
                const int h = 2 * i + halfsel;          // one head per 16-lane group
                const float* mr = &mbuf[h * M_STRIDE];
                float lg = 0.f;
                #pragma unroll 8
                for (int u = 0; u < UNIT_DIM; ++u) lg = fmaf(urow[u], mr[u], lg);
                lg = fmaf(qv, mr[UNIT_DIM], lg);        // appended q-value feature
                racc = fmaf(lg, lg, racc);

                float xs = lg * SCALE_F;
                if (act == 0) xs = MASK_F;
                float mx = xs;
                #pragma unroll
                for (int off = 8; off >= 1; off >>= 1)
                    mx = fmaxf(mx, __shfl_xor(mx, off, 16));
                const float ex = __expf(xs - mx);
                float sm = ex;
                #pragma unroll
                for (int off = 8; off >= 1; off >>= 1)
                    sm += __shfl_xor(sm, off, 16);
                const float p = ex / sm;

                float hq = qv * p;
                #pragma unroll
                for (int off = 8; off >= 1; off >>= 1)
                    hq += __shfl_xor(hq, off, 16);
                hqv[i] = hq;

                float et = p * __logf(p + 1e-8f);
                #pragma unroll
                for (int off = 8; off >= 1; off >>= 1)
                    et += __shfl_xor(et, off, 16);
                eacc[i] += et;
            }

            // q_tot[r] = sum_h |w_head| * head_q  (even heads lanes<16, odd lanes>=16)
            float ps = 0.f;
            #pragma unroll
            for (int i = 0; i < 4; ++i)
                ps = fmaf(sWh[rg * HEADS + 2 * i + halfsel], hqv[i], ps);
            ps += __shfl_xor(ps, 16, 32);
            if (lane == 0) out[b0 + rg] = ps;
        }
    }

    // deterministic block partials (no float atomics)
    #pragma unroll
    for (int off = 16; off >= 1; off >>= 1)
        racc += __shfl_xor(racc, off, 32);
    if (lane == 0) sRegW[wv] = racc;
    if ((lane & 15) == 0) {
        #pragma unroll
        for (int i = 0; i < 4; ++i)
            sEntW[wv * HEADS + 2 * i + halfsel] = eacc[i];
    }
    __syncthreads();
    if (tid < 9) {
        float s = 0.f;
        for (int w8 = 0; w8 < 8; ++w8)
            s += (tid == 0) ? sRegW[w8] : sEntW[w8 * HEADS + (tid - 1)];
        ws[blockIdx.x * 9 + tid] = s;
    }
}

__global__ void qatten_reduce(const float* __restrict__ ws, float* __restrict__ out)
{
    const int j = threadIdx.x;
    if (j < 9) {
        float s = 0.f;
        for (int b = 0; b < BSZ / ROWS; ++b) s += ws[b * 9 + j];   // fixed order
        if (j == 0)
            out[2 * BSZ] = REG_COEF * s / (float)(BSZ * N_AGENTS);
        else
            out[2 * BSZ + j] = -s / (float)BSZ;   // head_entropies[j-1]
    }
}

extern "C" void kernel_launch(void* const* d_in, const int* in_sizes, int n_in,
                              void* d_out, int out_size, void* d_ws, size_t ws_size,
                              hipStream_t stream)
{
    (void)in_sizes; (void)n_in; (void)out_size; (void)ws_size;
    const float* q_values = (const float*)d_in[0];
    const float* states   = (const float*)d_in[1];
    const int*   actions  = (const int*)  d_in[2];
    const float* W_sel    = (const float*)d_in[3];
    const float* W_key    = (const float*)d_in[4];
    const float* W_v1     = (const float*)d_in[5];
    const float* b_v1     = (const float*)d_in[6];
    const float* W_v2     = (const float*)d_in[7];
    const float* b_v2     = (const float*)d_in[8];
    const float* W_wh     = (const float*)d_in[9];
    const float* b_wh     = (const float*)d_in[10];
    float* out = (float*)d_out;
    float* ws  = (float*)d_ws;

    hipLaunchKernelGGL(qatten_fused, dim3(BSZ / ROWS), dim3(256), 0, stream,
                       q_values, states, actions, W_sel, W_key, W_v1, b_v1,
                       W_v2, b_v2, W_wh, b_wh, out, ws);
    hipLaunchKernelGGL(qatten_reduce, dim3(1), dim3(32), 0, stream, ws, out);
}